// ProtoInstanceAttention_90426241450027
// MI455X (gfx1250) — compile-verified
//
#include <hip/hip_runtime.h>
#include <math.h>

// ---------------------------------------------------------------------------
// ProtoInstanceAttention on gfx1250 (CDNA5, wave32, WMMA)
//
// Shapes: N_CLASSES=64, N_SUPPORT=5, N_QUERY=15, H=512
//   x: (1280, 512) f32   [row c*20+0..4 = support, c*20+5..19 = query]
//   W: (512, 512) f32, b: (512,) f32
// Output: dists (960, 64) f32
//
// Pipeline:
//   A   = x @ W^T + b                      (WMMA f32 16x16x4 GEMM, 1280x512x512)
//   SQ  = support_flat @ query^T           (WMMA f32 16x16x4 GEMM, 320x960x512)
//   G   = per-class support Gram (64x5x5), Qn = ||query||^2
//   scores[q,cs] = sum_h tanh(A[srow(cs),h] * A[qrow(q),h])   (v_tanh_f32)
//   dists[q,c]   = sum_ss' a a' G - 2 sum_s a SQ + Qn   (a = softmax(scores))
// ---------------------------------------------------------------------------

typedef float v2f __attribute__((ext_vector_type(2)));
typedef float v8f __attribute__((ext_vector_type(8)));

#define NCLS 64
#define NSUP 5
#define NQRY 15
#define HD   512
#define NSROW 320   // NCLS*NSUP
#define NQROW 960   // NCLS*NQRY

// Hardware tanh (TRANS op on gfx1250). Prefer the clang builtin; otherwise
// emit the instruction directly. The asm path appends a v_nop because the
// TRANS co-execution hazard (1 independent op before result use / source
// overwrite) is not visible to the compiler through inline asm.
__device__ __forceinline__ float hw_tanh(float x) {
#if defined(__has_builtin) && __has_builtin(__builtin_amdgcn_tanhf)
  return __builtin_amdgcn_tanhf(x);
#elif defined(__has_builtin) && __has_builtin(__builtin_amdgcn_tanh_f32)
  return __builtin_amdgcn_tanh_f32(x);
#else
  float r;
  asm volatile("v_tanh_f32 %0, %1\n\tv_nop" : "=v"(r) : "v"(x));
  return r;
#endif
}

__device__ __forceinline__ int support_row(int cs) { return (cs / NSUP) * 20 + (cs % NSUP); }
__device__ __forceinline__ int query_row(int q)    { return (q / NQRY) * 20 + NSUP + (q % NQRY); }

// ---------------------------------------------------------------------------
// scores[q][cs] = sum_h tanh(A[srow(cs)][h] * A[qrow(q)][h])
// (first in file so the disasm snippet shows its inner loop)
// 16q x 16cs tile per 256-thread block; h in two 256-chunks staged in LDS.
// Row stride 260 dwords: 260 % 64 == 4 -> float4 row reads hit all 64 banks;
// 260 % 4 == 0 keeps 16B alignment for float4.
// ---------------------------------------------------------------------------
#define HPAD 260
__global__ void __launch_bounds__(256) scores_kernel(
    const float* __restrict__ A, float* __restrict__ scores) {
  __shared__ float sS[16 * HPAD];
  __shared__ float sQ[16 * HPAD];

  const int cs0 = blockIdx.x << 4;                 // 0..304
  const int q0  = blockIdx.y << 4;                 // 0..944
  const int tid = threadIdx.x;
  const int lrow = tid >> 4;                       // loader row 0..15
  const int lcol = (tid & 15) << 4;                // loader col base (x16 floats)
  const int lcs  = tid & 15;                       // compute: local cs
  const int lq   = tid >> 4;                       // compute: local q

  const int sRowA = support_row(cs0 + lrow) * HD;
  const int qRowA = query_row(q0 + lrow) * HD;

  float acc = 0.f;
  for (int kc = 0; kc < HD; kc += 256) {
    const float4* __restrict__ gs = (const float4*)(A + sRowA + kc + lcol);
    const float4* __restrict__ gq = (const float4*)(A + qRowA + kc + lcol);
    float4* ds = (float4*)&sS[lrow * HPAD + lcol];
    float4* dq = (float4*)&sQ[lrow * HPAD + lcol];
#pragma unroll
    for (int v = 0; v < 4; ++v) { ds[v] = gs[v]; dq[v] = gq[v]; }
    __syncthreads();

    const float* __restrict__ ps = &sS[lcs * HPAD];
    const float* __restrict__ pq = &sQ[lq * HPAD];
#pragma unroll 4
    for (int h = 0; h < 256; h += 4) {
      const float4 sv = *(const float4*)(ps + h);
      const float4 qv = *(const float4*)(pq + h);
      acc += hw_tanh(sv.x * qv.x);
      acc += hw_tanh(sv.y * qv.y);
      acc += hw_tanh(sv.z * qv.z);
      acc += hw_tanh(sv.w * qv.w);
    }
    __syncthreads();
  }
  scores[(q0 + lq) * NSROW + cs0 + lcs] = acc;
}

// ---------------------------------------------------------------------------
// GEMM 1: A[m][n] = sum_k X[m][k] * W[n][k] + bias[n]   (M=1280, N=512, K=512)
// One wave per 16x16 output tile; V_WMMA_F32_16X16X4_F32, K-loop step 4.
// ISA 16x4 f32 A layout: lane L holds row M=L&15; VGPR j holds K = j + 2*(L>>4).
// ---------------------------------------------------------------------------
__global__ void __launch_bounds__(256) gemm_att_kernel(
    const float* __restrict__ X, const float* __restrict__ W,
    const float* __restrict__ bias, float* __restrict__ out) {
  const int lane = threadIdx.x & 31;
  const int wave = threadIdx.x >> 5;
  const int tile = blockIdx.x * 8 + wave;          // 2560 tiles = 80 x 32
  const int m0 = (tile >> 5) << 4;                 // tile/32 * 16
  const int n0 = (tile & 31) << 4;                 // tile%32 * 16
  const int half = lane >> 4;
  const int r16  = lane & 15;

  const float* __restrict__ xrow = X + (m0 + r16) * HD;
  const float* __restrict__ wrow = W + (n0 + r16) * HD;  // B[k][n] = W[n][k]

  v8f c = {};
  for (int k0 = 0; k0 < HD; k0 += 4) {
    const int k = k0 + 2 * half;
    v2f a  = { xrow[k], xrow[k + 1] };
    v2f bb = { wrow[k], wrow[k + 1] };
    c = __builtin_amdgcn_wmma_f32_16x16x4_f32(false, a, false, bb,
                                              (short)0, c, false, false);
  }

  const int col = n0 + r16;
  const float bv = bias[col];
#pragma unroll
  for (int r = 0; r < 8; ++r) {
    const int row = m0 + r + 8 * half;             // C/D: VGPR r -> M=r (+8 hi half)
    out[row * HD + col] = c[r] + bv;
  }
}

// ---------------------------------------------------------------------------
// GEMM 2: SQ[cs][q] = x[support_row(cs)] . x[query_row(q)]   (320 x 960, K=512)
// ---------------------------------------------------------------------------
__global__ void __launch_bounds__(256) gemm_sq_kernel(
    const float* __restrict__ X, float* __restrict__ SQ) {
  const int lane = threadIdx.x & 31;
  const int wave = threadIdx.x >> 5;
  const int tile = blockIdx.x * 8 + wave;          // 1200 tiles = 20 x 60
  const int cs0 = (tile % 20) << 4;
  const int q0  = (tile / 20) << 4;
  const int half = lane >> 4;
  const int r16  = lane & 15;

  const float* __restrict__ srow = X + support_row(cs0 + r16) * HD;
  const float* __restrict__ qrow = X + query_row(q0 + r16) * HD;

  v8f c = {};
  for (int k0 = 0; k0 < HD; k0 += 4) {
    const int k = k0 + 2 * half;
    v2f a  = { srow[k], srow[k + 1] };
    v2f bb = { qrow[k], qrow[k + 1] };
    c = __builtin_amdgcn_wmma_f32_16x16x4_f32(false, a, false, bb,
                                              (short)0, c, false, false);
  }

#pragma unroll
  for (int r = 0; r < 8; ++r) {
    const int row = cs0 + r + 8 * half;
    SQ[row * NQROW + q0 + r16] = c[r];
  }
}

// ---------------------------------------------------------------------------
// Gram: G[c][s][s'] = support[c,s] . support[c,s']  (64*25 dot products)
// ---------------------------------------------------------------------------
__global__ void gram_kernel(const float* __restrict__ X, float* __restrict__ G) {
  const int t = blockIdx.x * blockDim.x + threadIdx.x;
  if (t >= NCLS * 25) return;
  const int c = t / 25, p = t % 25, s = p / 5, sp = p % 5;
  const float* __restrict__ a = X + (c * 20 + s) * HD;
  const float* __restrict__ b = X + (c * 20 + sp) * HD;
  float acc = 0.f;
  for (int h = 0; h < HD; ++h) acc += a[h] * b[h];
  G[t] = acc;
}

// Qn[q] = ||query_q||^2
__global__ void qnorm_kernel(const float* __restrict__ X, float* __restrict__ Qn) {
  const int q = blockIdx.x * blockDim.x + threadIdx.x;
  if (q >= NQROW) return;
  const float* __restrict__ a = X + query_row(q) * HD;
  float acc = 0.f;
  for (int h = 0; h < HD; ++h) acc += a[h] * a[h];
  Qn[q] = acc;
}

// ---------------------------------------------------------------------------
// dists[q][c] = sum_{s,s'} a_s a_s' G[c,s,s'] - 2 sum_s a_s SQ[c*5+s][q] + Qn[q]
// where a = softmax(scores[q, c*5 .. c*5+4])
// ---------------------------------------------------------------------------
__global__ void dists_kernel(const float* __restrict__ scores,
                             const float* __restrict__ SQ,
                             const float* __restrict__ G,
                             const float* __restrict__ Qn,
                             float* __restrict__ out) {
  const int t = blockIdx.x * blockDim.x + threadIdx.x;
  if (t >= NQROW * NCLS) return;
  const int q = t >> 6;                            // t / 64
  const int c = t & 63;

  float s[NSUP], a[NSUP];
  float m = -1e30f;
#pragma unroll
  for (int i = 0; i < NSUP; ++i) {
    s[i] = scores[q * NSROW + c * NSUP + i];
    m = fmaxf(m, s[i]);
  }
  float sum = 0.f;
#pragma unroll
  for (int i = 0; i < NSUP; ++i) { a[i] = __expf(s[i] - m); sum += a[i]; }
  const float inv = 1.f / sum;
#pragma unroll
  for (int i = 0; i < NSUP; ++i) a[i] *= inv;

  float acc = Qn[q];
  const float* __restrict__ g = G + c * 25;
#pragma unroll
  for (int i = 0; i < NSUP; ++i) {
    acc -= 2.f * a[i] * SQ[(c * NSUP + i) * NQROW + q];
#pragma unroll
    for (int j = 0; j < NSUP; ++j) acc += a[i] * a[j] * g[i * 5 + j];
  }
  out[t] = acc;                                    // out[q*64 + c]
}

// ---------------------------------------------------------------------------
extern "C" void kernel_launch(void* const* d_in, const int* in_sizes, int n_in,
                              void* d_out, int out_size, void* d_ws, size_t ws_size,
                              hipStream_t stream) {
  const float* x    = (const float*)d_in[0];   // (1280, 512)
  const float* W    = (const float*)d_in[1];   // (512, 512)
  const float* bias = (const float*)d_in[2];   // (512,)
  // d_in[3] = y (identity permutation after argsort), d_in[4..7] scalars: unused
  float* out = (float*)d_out;                  // (960, 64)

  // Workspace layout (floats)
  float* ws = (float*)d_ws;
  float* A      = ws;                          // 1280*512   = 655360
  float* SQ     = A + 1280 * HD;               // 320*960    = 307200
  float* scores = SQ + NSROW * NQROW;          // 960*320    = 307200
  float* G      = scores + NQROW * NSROW;      // 64*25      = 1600
  float* Qn     = G + NCLS * 25;               // 960
  (void)ws_size; (void)in_sizes; (void)n_in; (void)out_size;

  // 1) Attended features: A = x @ W^T + b   (2560 tiles, 8 waves/block)
  gemm_att_kernel<<<320, 256, 0, stream>>>(x, W, bias, A);

  // 2) SQ = support . query^T               (1200 tiles)
  gemm_sq_kernel<<<150, 256, 0, stream>>>(x, SQ);

  // 3) Per-class support Gram + query norms
  gram_kernel<<<(NCLS * 25 + 255) / 256, 256, 0, stream>>>(x, G);
  qnorm_kernel<<<(NQROW + 255) / 256, 256, 0, stream>>>(x, Qn);

  // 4) tanh scores (dominant cost: 157M v_tanh_f32)
  dim3 sgrid(NSROW / 16, NQROW / 16);          // (20, 60)
  scores_kernel<<<sgrid, 256, 0, stream>>>(A, scores);

  // 5) softmax + quadratic-form distances
  dists_kernel<<<(NQROW * NCLS + 255) / 256, 256, 0, stream>>>(scores, SQ, G, Qn, out);
}